// GroupWiseContrastiveLoss_42021960024483
// MI455X (gfx1250) — compile-verified
//
#include <hip/hip_runtime.h>

#define N_VIDEOS 256
#define EMBED_DIM 1024

typedef __attribute__((ext_vector_type(2))) float v2f;
typedef __attribute__((ext_vector_type(8))) float v8f;

// ---------------------------------------------------------------------------
// Kernel 1: segment-sum pooling.  One block per video; 256 threads each own
// 4 contiguous columns (float4) and accumulate over the video's rows.
// Streams the full 16.7 MB input once -> bandwidth-bound (roofline floor).
// ---------------------------------------------------------------------------
__global__ void pool_segments_kernel(const float* __restrict__ x,
                                     const int* __restrict__ seg_sizes,
                                     float* __restrict__ out) {
    const int vid = blockIdx.x;

    // Prefix sum over (uniform) segment sizes -> scalar loop, SALU-only.
    int start = 0;
    for (int i = 0; i < vid; ++i) start += seg_sizes[i];
    const int cnt = seg_sizes[vid];

    const int col = threadIdx.x * 4;                 // 256 thr * 4 = 1024 cols
    const float* p = x + (size_t)start * EMBED_DIM + col;

    float4 acc = make_float4(0.f, 0.f, 0.f, 0.f);
    for (int r = 0; r < cnt; ++r) {
        float4 v = *(const float4*)p;
        acc.x += v.x; acc.y += v.y; acc.z += v.z; acc.w += v.w;
        p += EMBED_DIM;
    }
    *(float4*)(out + (size_t)vid * EMBED_DIM + col) = acc;
}

// ---------------------------------------------------------------------------
// Kernel 2: blk = A (256x1024) * B^T (1024x256) via V_WMMA_F32_16X16X4_F32.
// One wave32 per 16x16 output tile (EXEC all ones, as WMMA requires).
// A-frag (16x4) and B-frag (4x16) share the same lane addressing for
// row-major A,B: lane L reads float2 at row tile+(L&15), col k+(L>>4)*2.
// ---------------------------------------------------------------------------
__global__ void gemm16_wmma_f32_kernel(const float* __restrict__ A,
                                       const float* __restrict__ B,
                                       float* __restrict__ C) {
    const int tm   = blockIdx.x * 16;
    const int tn   = blockIdx.y * 16;
    const int lane = threadIdx.x;            // 0..31 (wave32)
    const int row  = lane & 15;
    const int kofs = (lane >> 4) * 2;        // 0 for lanes 0-15, 2 for 16-31

    const float* ap = A + (size_t)(tm + row) * EMBED_DIM + kofs;
    const float* bp = B + (size_t)(tn + row) * EMBED_DIM + kofs;

    v8f acc = {};
#pragma unroll 8
    for (int k = 0; k < EMBED_DIM; k += 4) {
        v2f a = *(const v2f*)(ap + k);
        v2f b = *(const v2f*)(bp + k);
        // 8 args: (neg_a, A, neg_b, B, c_mod, C, reuse_a, reuse_b)
        acc = __builtin_amdgcn_wmma_f32_16x16x4_f32(
            false, a, false, b, (short)0, acc, false, false);
    }

    // C/D layout: VGPR r -> M = r + (lane<16 ? 0 : 8), N = lane&15
    const int mbase = (lane >> 4) * 8;
#pragma unroll
    for (int r = 0; r < 8; ++r) {
        C[(size_t)(tm + mbase + r) * N_VIDEOS + (tn + row)] = acc[r];
    }
}

// ---------------------------------------------------------------------------
// Kernel 3: hinge loss.  S = blk / (nc_i * ncap_j); MARGIN = 0.
// loss = sum_{i!=j} max(S - diag_i, 0) + max(S - diag_j, 0).
// Single 256-thread block; 256 KB of blk, trivial.
// ---------------------------------------------------------------------------
__global__ void hinge_loss_kernel(const float* __restrict__ blk,
                                  const int* __restrict__ num_clips,
                                  const int* __restrict__ num_caps,
                                  float* __restrict__ out) {
    __shared__ float diag[N_VIDEOS];
    __shared__ float partial[N_VIDEOS];

    const int i = threadIdx.x;
    const float nci = (float)num_clips[i];
    diag[i] = blk[(size_t)i * N_VIDEOS + i] / (nci * (float)num_caps[i]);
    __syncthreads();

    const float di = diag[i];
    float sum = 0.f;
    for (int j = 0; j < N_VIDEOS; ++j) {
        if (j == i) continue;
        const float S = blk[(size_t)i * N_VIDEOS + j] / (nci * (float)num_caps[j]);
        const float cs = S - di;       // caption retrieval: S - diag[i]
        const float ci = S - diag[j];  // image retrieval:   S - diag[j]
        if (cs > 0.f) sum += cs;
        if (ci > 0.f) sum += ci;
    }
    partial[i] = sum;
    __syncthreads();

    for (int sft = N_VIDEOS / 2; sft > 0; sft >>= 1) {
        if (i < sft) partial[i] += partial[i + sft];
        __syncthreads();
    }
    if (i == 0) out[0] = partial[0];
}

// ---------------------------------------------------------------------------
extern "C" void kernel_launch(void* const* d_in, const int* in_sizes, int n_in,
                              void* d_out, int out_size, void* d_ws, size_t ws_size,
                              hipStream_t stream) {
    const float* im        = (const float*)d_in[0];   // [total_clips, 1024]
    const float* s         = (const float*)d_in[1];   // [total_caps, 1024]
    const int*   num_clips = (const int*)d_in[2];     // [256]
    const int*   num_caps  = (const int*)d_in[3];     // [256]

    float* A   = (float*)d_ws;                        // 256*1024 f32 = 1 MB
    float* B   = A + N_VIDEOS * EMBED_DIM;            // 1 MB
    float* blk = B + N_VIDEOS * EMBED_DIM;            // 256*256 f32 = 256 KB

    pool_segments_kernel<<<N_VIDEOS, 256, 0, stream>>>(im, num_clips, A);
    pool_segments_kernel<<<N_VIDEOS, 256, 0, stream>>>(s,  num_caps,  B);

    dim3 grid(N_VIDEOS / 16, N_VIDEOS / 16);          // 16x16 tiles of 256x256
    gemm16_wmma_f32_kernel<<<grid, 32, 0, stream>>>(A, B, blk);

    hinge_loss_kernel<<<1, N_VIDEOS, 0, stream>>>(blk, num_clips, num_caps,
                                                  (float*)d_out);
}